// MLPBuilder_22754736734851
// MI455X (gfx1250) — compile-verified
//
#include <hip/hip_runtime.h>
#include <hip/hip_bf16.h>

typedef __attribute__((ext_vector_type(2))) float v2f;
typedef __attribute__((ext_vector_type(8))) float v8f;

#define NROWS 1024
#define DDIM  128
#define HDIM  256

// ---------------------------------------------------------------------------
// Phase 1: la = x @ W1[:, :D]^T  and  lb = x @ W1[:, D:]^T via fp32 WMMA.
// One wave (32 lanes) computes one 16x16 output tile, K-loop of 32 WMMAs.
//   blockIdx.x: i-tile (64), blockIdx.y: h-tile (16), blockIdx.z: {la, lb}
// ---------------------------------------------------------------------------
__global__ __launch_bounds__(32)
void mlp_gemm1_wmma(const float* __restrict__ x,   // [1024,128]
                    const float* __restrict__ W1,  // [256,256] row-major
                    float* __restrict__ la,        // [1024,256]
                    float* __restrict__ lb)        // [1024,256]
{
    const int lane = threadIdx.x;        // 0..31
    const int half = lane >> 4;          // 0 or 1 (K sub-pair select)
    const int l16  = lane & 15;

    const int i0 = blockIdx.x * 16;
    const int h0 = blockIdx.y * 16;
    const int colofs = blockIdx.z ? DDIM : 0;
    float* __restrict__ out = blockIdx.z ? lb : la;

    // A: x row (M = i0+l16), K pairs at k + 2*half
    const float* __restrict__ arow = x  + (i0 + l16) * DDIM + 2 * half;
    // B: B[k][n] = W1[h0+n][colofs+k] -> lane n walks its W1 row
    const float* __restrict__ brow = W1 + (h0 + l16) * (2 * DDIM) + colofs + 2 * half;

    v8f acc = {};
#pragma unroll
    for (int k = 0; k < DDIM; k += 4) {
        v2f a = *(const v2f*)(arow + k);
        v2f b = *(const v2f*)(brow + k);
        acc = __builtin_amdgcn_wmma_f32_16x16x4_f32(
            /*neg_a=*/false, a, /*neg_b=*/false, b,
            /*c_mod=*/(short)0, acc, /*reuse_a=*/false, /*reuse_b=*/false);
    }

    // D layout: VGPR r -> row i0 + r + 8*half, col h0 + l16
#pragma unroll
    for (int r = 0; r < 8; ++r) {
        out[(i0 + r + 8 * half) * HDIM + h0 + l16] = acc[r];
    }
}

// ---------------------------------------------------------------------------
// Phase 2: adj[i,j] = (sum_h relu(la[i,h]+lb[j,h]+b1[h]) * (W2[1,h]-W2[0,h])
//                      + (b2[1]-b2[0])) > 0, diagonal forced to 1.
// One block per (i, 256-wide j chunk); t = la[i]+b1 and w = W2[1]-W2[0] in LDS.
// ---------------------------------------------------------------------------
__global__ __launch_bounds__(256)
void mlp_adj(const float* __restrict__ la,   // [1024,256]
             const float* __restrict__ lb,   // [1024,256]
             const float* __restrict__ b1,   // [256]
             const float* __restrict__ W2,   // [2,256]
             const float* __restrict__ b2,   // [2]
             float* __restrict__ adj)        // [1024,1024]
{
    __shared__ float t[HDIM];   // la[i,:] + b1
    __shared__ float w[HDIM];   // W2[1,:] - W2[0,:]

    const int i   = blockIdx.x;
    const int tid = threadIdx.x;

    t[tid] = la[i * HDIM + tid] + b1[tid];
    w[tid] = W2[HDIM + tid] - W2[tid];
    __syncthreads();

    const int j = blockIdx.y * 256 + tid;
    const float4* __restrict__ lbr = (const float4*)(lb + j * HDIM);

    float s = b2[1] - b2[0];
#pragma unroll 8
    for (int h4 = 0; h4 < HDIM / 4; ++h4) {
        const float4 v  = lbr[h4];
        const float4 tv = *(const float4*)&t[h4 * 4];
        const float4 wv = *(const float4*)&w[h4 * 4];
        s = fmaf(fmaxf(tv.x + v.x, 0.0f), wv.x, s);
        s = fmaf(fmaxf(tv.y + v.y, 0.0f), wv.y, s);
        s = fmaf(fmaxf(tv.z + v.z, 0.0f), wv.z, s);
        s = fmaf(fmaxf(tv.w + v.w, 0.0f), wv.w, s);
    }

    adj[i * NROWS + j] = (i == j) ? 1.0f : (s > 0.0f ? 1.0f : 0.0f);
}

// ---------------------------------------------------------------------------
// Launch: inputs in setup_inputs() order: x, W1, b1, W2, b2.
// Workspace: la (1 MB) then lb (1 MB).
// ---------------------------------------------------------------------------
extern "C" void kernel_launch(void* const* d_in, const int* in_sizes, int n_in,
                              void* d_out, int out_size, void* d_ws, size_t ws_size,
                              hipStream_t stream) {
    const float* x  = (const float*)d_in[0];
    const float* W1 = (const float*)d_in[1];
    const float* b1 = (const float*)d_in[2];
    const float* W2 = (const float*)d_in[3];
    const float* b2 = (const float*)d_in[4];
    float* adj = (float*)d_out;

    float* la = (float*)d_ws;
    float* lb = la + NROWS * HDIM;

    dim3 g1(NROWS / 16, HDIM / 16, 2);
    mlp_gemm1_wmma<<<g1, 32, 0, stream>>>(x, W1, la, lb);

    dim3 g2(NROWS, NROWS / 256);
    mlp_adj<<<g2, 256, 0, stream>>>(la, lb, b1, W2, b2, adj);
}